// Block_89172110999857
// MI455X (gfx1250) — compile-verified
//
#include <hip/hip_runtime.h>
#include <hip/hip_bf16.h>
#include <cstddef>

// Problem constants (from reference): B=32, T=2048, H=16, HD=128, C=2048
constexpr int Bb = 32;
constexpr int Tt = 2048;
constexpr int Hh = 16;
constexpr int HD = 128;
constexpr int Cc = 2048;

typedef __attribute__((ext_vector_type(2))) float v2f;
typedef __attribute__((ext_vector_type(8))) float v8f;

// ---------------------------------------------------------------------------
// LayerNorm: one block per row of (B, C). 256 threads, 8 elems/thread.
// ---------------------------------------------------------------------------
__global__ __launch_bounds__(256) void ln_kernel(const float* __restrict__ x,
                                                 const float* __restrict__ scale,
                                                 const float* __restrict__ bias,
                                                 float* __restrict__ out) {
  __shared__ float red[256];
  const int b = blockIdx.x;
  const int tid = threadIdx.x;
  const float* row = x + (size_t)b * Cc;

  float v[8];
  float s = 0.f;
#pragma unroll
  for (int i = 0; i < 8; ++i) {
    v[i] = row[tid + i * 256];
    s += v[i];
  }
  red[tid] = s;
  __syncthreads();
  for (int off = 128; off > 0; off >>= 1) {
    if (tid < off) red[tid] += red[tid + off];
    __syncthreads();
  }
  const float mean = red[0] * (1.0f / Cc);
  __syncthreads();

  float s2 = 0.f;
#pragma unroll
  for (int i = 0; i < 8; ++i) {
    float d = v[i] - mean;
    s2 += d * d;
  }
  red[tid] = s2;
  __syncthreads();
  for (int off = 128; off > 0; off >>= 1) {
    if (tid < off) red[tid] += red[tid + off];
    __syncthreads();
  }
  const float inv = rsqrtf(red[0] * (1.0f / Cc) + 1e-5f);
  __syncthreads();

#pragma unroll
  for (int i = 0; i < 8; ++i) {
    int col = tid + i * 256;
    out[(size_t)b * Cc + col] = (v[i] - mean) * inv * scale[col] + bias[col];
  }
}

// ---------------------------------------------------------------------------
// WMMA fp32 GEMM: out(M=32, N) = A(32, K) @ W(K, N) + bias, optional epilogue.
// One wave computes one 16x16 tile with V_WMMA_F32_16X16X4_F32, stepping K by 4.
// K, N are compile-time so W-row loads use immediate offsets and the K-loop is
// pointer-bumped (2 pointer adds per 8 WMMAs). Block = 128 threads (4 waves,
// 4 adjacent N-tiles). grid = (N/64, M/16).
// EPI: 0 = bias only, 1 = bias + residual add, 2 = bias + tanh-GELU.
// ---------------------------------------------------------------------------
template <int EPI, int K, int N>
__global__ __launch_bounds__(128) void gemm_wmma_kernel(
    const float* __restrict__ A, const float* __restrict__ W,
    const float* __restrict__ bias, const float* __restrict__ resid,
    float* __restrict__ out) {
  constexpr int KSTEP = 32;  // 8 wmma per group
  const int lane = threadIdx.x & 31;
  const int wave = threadIdx.x >> 5;
  const int l = lane & 15;
  const int half = lane >> 4;  // 0: lanes 0-15, 1: lanes 16-31
  const int nBase = (blockIdx.x * 4 + wave) << 4;
  const int mBase = blockIdx.y << 4;
  const int n = nBase + l;
  const int m = mBase + l;

  // A fragment (16x4, 32-bit): lane l holds row M=l; half 0 -> K,K+1, half 1 -> K+2,K+3
  const float* __restrict__ ap = A + (size_t)m * K + 2 * half;
  // B fragment (4x16): VGPR0 holds rows K + (half?2:0), VGPR1 rows K+1 / K+3, col N=l
  const float* __restrict__ wp = W + (size_t)(2 * half) * N + n;

  v8f c = {0.f, 0.f, 0.f, 0.f, 0.f, 0.f, 0.f, 0.f};

#pragma unroll 1
  for (int k = 0; k < K; k += KSTEP) {
    // Speculative prefetch of the weight stream one group ahead
    // (safe past end of buffer: dropped on translation failure).
    __builtin_prefetch(wp + (size_t)KSTEP * N, 0, 1);
#pragma unroll
    for (int u = 0; u < KSTEP / 4; ++u) {
      float2 av = *reinterpret_cast<const float2*>(ap + u * 4);
      v2f a;
      a.x = av.x;
      a.y = av.y;
      v2f bfr;
      bfr.x = wp[(size_t)(u * 4) * N];        // compile-time immediate offset
      bfr.y = wp[(size_t)(u * 4) * N + N];
      // D = A*B + C  (8 args: neg_a, A, neg_b, B, c_mod, C, reuse_a, reuse_b)
      c = __builtin_amdgcn_wmma_f32_16x16x4_f32(false, a, false, bfr, (short)0,
                                                c, false, false);
    }
    ap += KSTEP;
    wp += (size_t)KSTEP * N;
  }

  const float bv = bias[n];
#pragma unroll
  for (int v = 0; v < 8; ++v) {
    // C/D layout: VGPR v, lanes 0-15 -> M = v, lanes 16-31 -> M = v + 8
    int row = mBase + v + 8 * half;
    float val = c[v] + bv;
    if (EPI == 1) val += resid[(size_t)row * N + n];
    if (EPI == 2) {
      float xx = val;
      float t = tanhf(0.7978845608028654f * (xx + 0.044715f * xx * xx * xx));
      val = 0.5f * xx * (1.0f + t);
    }
    out[(size_t)row * N + n] = val;
  }
}

// ---------------------------------------------------------------------------
// Attention: one block per (b, h). Scores staged in LDS, 2-phase softmax,
// coalesced weighted sum over prev_v.
// qkv layout: (B, 3C); q for head h = qkv[b, h*HD : h*HD+HD].
// prev_k/prev_v layout: (B, T, H, HD).
// ---------------------------------------------------------------------------
__global__ __launch_bounds__(256) void attn_kernel(const float* __restrict__ qkv,
                                                   const float* __restrict__ pk,
                                                   const float* __restrict__ pv,
                                                   const int* __restrict__ pos,
                                                   float* __restrict__ out) {
  __shared__ float sc[Tt];   // 8 KB scores
  __shared__ float qs[HD];   // query vector
  __shared__ float red[256]; // reductions / partial sums
  const int bh = blockIdx.x;
  const int b = bh >> 4;
  const int h = bh & 15;
  const int tid = threadIdx.x;
  const int P = pos[b];  // valid t: [0, P]

  if (tid < HD) qs[tid] = qkv[(size_t)b * (3 * Cc) + h * HD + tid];
  __syncthreads();

  const float scl = 0.08838834764831845f;  // 1/sqrt(128)
  float lmax = -3.4e38f;
  for (int t = tid; t < Tt; t += 256) {
    float s = -3.4e38f;
    if (t <= P) {
      const float4* kr =
          reinterpret_cast<const float4*>(pk + (((size_t)b * Tt + t) * Hh + h) * HD);
      const float4* qq4 = reinterpret_cast<const float4*>(qs);
      float acc = 0.f;
#pragma unroll
      for (int d = 0; d < HD / 4; ++d) {
        float4 kk = kr[d];
        float4 qq = qq4[d];
        acc += qq.x * kk.x + qq.y * kk.y + qq.z * kk.z + qq.w * kk.w;
      }
      s = acc * scl;
    }
    sc[t] = s;
    lmax = fmaxf(lmax, s);
  }
  red[tid] = lmax;
  __syncthreads();
  for (int off = 128; off > 0; off >>= 1) {
    if (tid < off) red[tid] = fmaxf(red[tid], red[tid + off]);
    __syncthreads();
  }
  const float gmax = red[0];
  __syncthreads();

  float lsum = 0.f;
  for (int t = tid; t < Tt; t += 256) {
    float e = (t <= P) ? __expf(sc[t] - gmax) : 0.f;
    sc[t] = e;
    lsum += e;
  }
  red[tid] = lsum;
  __syncthreads();
  for (int off = 128; off > 0; off >>= 1) {
    if (tid < off) red[tid] += red[tid + off];
    __syncthreads();
  }
  const float inv = 1.0f / red[0];
  __syncthreads();

  // Weighted sum over prev_v: threads 0..127 own dim d, split t-range in two.
  const int d = tid & 127;
  const int hsel = tid >> 7;
  float acc = 0.f;
  for (int t = hsel; t <= P; t += 2) {
    acc += sc[t] * pv[(((size_t)b * Tt + t) * Hh + h) * HD + d];
  }
  red[tid] = acc;
  __syncthreads();
  if (tid < 128) {
    out[(size_t)b * Cc + h * HD + d] = (red[tid] + red[tid + 128]) * inv;
  }
}

// ---------------------------------------------------------------------------
// Extract k_new / v_new from the qkv buffer.
// ---------------------------------------------------------------------------
__global__ __launch_bounds__(256) void kv_out_kernel(const float* __restrict__ qkv,
                                                     float* __restrict__ kout,
                                                     float* __restrict__ vout) {
  const int i = blockIdx.x * 256 + threadIdx.x;  // [0, B*C)
  const int b = i / Cc;
  const int c = i % Cc;
  kout[i] = qkv[(size_t)b * (3 * Cc) + Cc + c];
  vout[i] = qkv[(size_t)b * (3 * Cc) + 2 * Cc + c];
}

// ---------------------------------------------------------------------------
extern "C" void kernel_launch(void* const* d_in, const int* in_sizes, int n_in,
                              void* d_out, int out_size, void* d_ws, size_t ws_size,
                              hipStream_t stream) {
  const float* x      = (const float*)d_in[0];
  const float* prev_k = (const float*)d_in[1];
  const float* prev_v = (const float*)d_in[2];
  const int*   pos    = (const int*)d_in[3];
  const float* ln1_s  = (const float*)d_in[4];
  const float* ln1_b  = (const float*)d_in[5];
  const float* w_attn = (const float*)d_in[6];
  const float* b_attn = (const float*)d_in[7];
  const float* w_proj = (const float*)d_in[8];
  const float* b_proj = (const float*)d_in[9];
  const float* ln2_s  = (const float*)d_in[10];
  const float* ln2_b  = (const float*)d_in[11];
  const float* w_fc   = (const float*)d_in[12];
  const float* b_fc   = (const float*)d_in[13];
  const float* w_mlp  = (const float*)d_in[14];
  const float* b_mlp  = (const float*)d_in[15];

  float* ws     = (float*)d_ws;
  float* h      = ws;                  // 65536
  float* qkv    = h + 65536;           // 196608
  float* attn_o = qkv + 196608;        // 65536
  float* x1     = attn_o + 65536;      // 65536
  float* h2     = x1 + 65536;          // 65536
  float* fcact  = h2 + 65536;          // 262144
  // total ~2.75 MB of f32 workspace

  float* out_x = (float*)d_out;        // 65536
  float* out_k = out_x + 65536;        // 65536
  float* out_v = out_k + 65536;        // 65536

  // 1) h = LN1(x)
  ln_kernel<<<Bb, 256, 0, stream>>>(x, ln1_s, ln1_b, h);
  // 2) qkv = h @ w_attn + b_attn          (32 x 6144)
  gemm_wmma_kernel<0, Cc, 3 * Cc><<<dim3((3 * Cc) / 64, 2), 128, 0, stream>>>(
      h, w_attn, b_attn, nullptr, qkv);
  // 3) k_new / v_new outputs
  kv_out_kernel<<<(Bb * Cc) / 256, 256, 0, stream>>>(qkv, out_k, out_v);
  // 4) attention -> attn_o (B, C)
  attn_kernel<<<Bb * Hh, 256, 0, stream>>>(qkv, prev_k, prev_v, pos, attn_o);
  // 5) x1 = x + attn_o @ w_attn_proj + b  (32 x 2048, residual epilogue)
  gemm_wmma_kernel<1, Cc, Cc><<<dim3(Cc / 64, 2), 128, 0, stream>>>(
      attn_o, w_proj, b_proj, x, x1);
  // 6) h2 = LN2(x1)
  ln_kernel<<<Bb, 256, 0, stream>>>(x1, ln2_s, ln2_b, h2);
  // 7) fcact = gelu(h2 @ w_fc + b_fc)     (32 x 8192, GELU epilogue)
  gemm_wmma_kernel<2, Cc, 4 * Cc><<<dim3((4 * Cc) / 64, 2), 128, 0, stream>>>(
      h2, w_fc, b_fc, nullptr, fcact);
  // 8) out_x = x1 + fcact @ w_mlp + b     (32 x 2048, residual epilogue)
  gemm_wmma_kernel<1, 4 * Cc, Cc><<<dim3(Cc / 64, 2), 128, 0, stream>>>(
      fcact, w_mlp, b_mlp, x1, out_x);
}